// GCN_38938173505666
// MI455X (gfx1250) — compile-verified
//
#include <hip/hip_runtime.h>
#include <hip/hip_bf16.h>
#include <math.h>

#define N_NODES 10000
#define N_EDGES 320000
#define F_IN    256
#define F_HID   256
#define F_OUT   16
#define BN_EPS  1e-5f

typedef __attribute__((ext_vector_type(2))) float v2f;
typedef __attribute__((ext_vector_type(8))) float v8f;

// ---------------------------------------------------------------------------
// WMMA helper: D = A(16x4 f32) * B(4x16 f32) + C(16x16 f32)
// 8-arg pattern: (neg_a, A, neg_b, B, c_mod, C, reuse_a, reuse_b)
// ---------------------------------------------------------------------------
__device__ __forceinline__ v8f wmma4(v2f a, v2f b, v8f c) {
    return __builtin_amdgcn_wmma_f32_16x16x4_f32(
        false, a, false, b, (short)0, c, false, false);
}

// A fragment: 16x4 tile at (m0,k0) of row-major A (ld = lda).
// Lanes 0-15: row = lane, K = k0+0 / k0+1 ; lanes 16-31: K = k0+2 / k0+3.
__device__ __forceinline__ v2f load_a(const float* A, int lda, int m0, int k0, int lane) {
    const int half = lane >> 4;          // 0 or 1
    const int row  = lane & 15;
    const float* p = A + (size_t)(m0 + row) * lda + k0 + (half << 1);
    v2f r; r.x = p[0]; r.y = p[1];
    return r;
}

// B fragment: 4x16 tile at (k0,n0) of row-major B (ld = ldb).
// Lanes 0-15: col = lane, K = k0+0 / k0+1 ; lanes 16-31: K = k0+2 / k0+3.
__device__ __forceinline__ v2f load_b(const float* B, int ldb, int k0, int n0, int lane) {
    const int half = lane >> 4;
    const int col  = lane & 15;
    const float* p = B + (size_t)(k0 + (half << 1)) * ldb + n0 + col;
    v2f r; r.x = p[0]; r.y = p[ldb];
    return r;
}

// C/D 16x16: VGPR r holds row r (lanes 0-15, N=lane) and row r+8 (lanes 16-31).
__device__ __forceinline__ void store_c(float* C, int ldc, int m0, int n0, int lane, v8f acc) {
    const int half = lane >> 4;
    const int col  = lane & 15;
#pragma unroll
    for (int r = 0; r < 8; ++r)
        C[(size_t)(m0 + r + half * 8) * ldc + n0 + col] = acc[r];
}

// ---------------------------------------------------------------------------
// Graph kernels
// ---------------------------------------------------------------------------
__global__ void deg_accum_kernel(const int* __restrict__ dst, float* __restrict__ deg) {
    int e = blockIdx.x * blockDim.x + threadIdx.x;
    if (e < N_EDGES) atomicAdd(&deg[dst[e]], 1.0f);
}

__global__ void deg_inv_kernel(float* __restrict__ deg) {
    int n = blockIdx.x * blockDim.x + threadIdx.x;
    if (n < N_NODES) deg[n] = 1.0f / fmaxf(deg[n], 1.0f);
}

// agg1[dst] += x[src], 4 floats per thread (64 chunks per edge)
__global__ void agg1_scatter_kernel(const float* __restrict__ x,
                                    const int* __restrict__ src,
                                    const int* __restrict__ dst,
                                    float* __restrict__ agg1) {
    long long g = (long long)blockIdx.x * blockDim.x + threadIdx.x;
    if (g >= (long long)N_EDGES * 64) return;
    int e  = (int)(g >> 6);
    int c4 = ((int)g & 63) << 2;
    const float4 v = *(const float4*)(x + (size_t)src[e] * F_IN + c4);
    float* a = agg1 + (size_t)dst[e] * F_IN + c4;
    atomicAdd(a + 0, v.x);
    atomicAdd(a + 1, v.y);
    atomicAdd(a + 2, v.z);
    atomicAdd(a + 3, v.w);
}

template <int COLS>
__global__ void scale_rows_kernel(float* __restrict__ m, const float* __restrict__ deg_inv,
                                  long long total) {
    long long g = (long long)blockIdx.x * blockDim.x + threadIdx.x;
    if (g < total) m[g] *= deg_inv[g / COLS];   // COLS is a power of two -> shift
}

// ---------------------------------------------------------------------------
// GEMM1 (fused): C = A1 @ B1 + A2 @ B2   (10000x256 @ 256x256, twice)
// One wave per 16x64 output tile: one A fragment feeds 4 WMMAs.
// 625 M-tiles x 4 N-groups = 2500 waves.
// ---------------------------------------------------------------------------
__global__ __launch_bounds__(256) void gemm1_kernel(const float* __restrict__ A1,
                                                    const float* __restrict__ B1,
                                                    const float* __restrict__ A2,
                                                    const float* __restrict__ B2,
                                                    float* __restrict__ C) {
    const int lane = threadIdx.x & 31;
    const int wave = threadIdx.x >> 5;
    const int tile = blockIdx.x * 8 + wave;   // 313 blocks * 8 waves >= 2500
    if (tile >= 2500) return;                 // per-wave uniform exit; EXEC stays full
    const int mt = tile >> 2;                 // 0..624
    const int ng = tile & 3;                  // 0..3
    const int m0 = mt * 16, n0 = ng * 64;

    v8f acc0 = {}, acc1 = {}, acc2 = {}, acc3 = {};
#pragma unroll 2
    for (int k0 = 0; k0 < F_IN; k0 += 4) {
        v2f a = load_a(A1, F_IN, m0, k0, lane);
        acc0 = wmma4(a, load_b(B1, F_HID, k0, n0 +  0, lane), acc0);
        acc1 = wmma4(a, load_b(B1, F_HID, k0, n0 + 16, lane), acc1);
        acc2 = wmma4(a, load_b(B1, F_HID, k0, n0 + 32, lane), acc2);
        acc3 = wmma4(a, load_b(B1, F_HID, k0, n0 + 48, lane), acc3);
    }
#pragma unroll 2
    for (int k0 = 0; k0 < F_IN; k0 += 4) {
        v2f a = load_a(A2, F_IN, m0, k0, lane);
        acc0 = wmma4(a, load_b(B2, F_HID, k0, n0 +  0, lane), acc0);
        acc1 = wmma4(a, load_b(B2, F_HID, k0, n0 + 16, lane), acc1);
        acc2 = wmma4(a, load_b(B2, F_HID, k0, n0 + 32, lane), acc2);
        acc3 = wmma4(a, load_b(B2, F_HID, k0, n0 + 48, lane), acc3);
    }
    store_c(C, F_HID, m0, n0 +  0, lane, acc0);
    store_c(C, F_HID, m0, n0 + 16, lane, acc1);
    store_c(C, F_HID, m0, n0 + 32, lane, acc2);
    store_c(C, F_HID, m0, n0 + 48, lane, acc3);
}

// ---------------------------------------------------------------------------
// GEMM3 (dual output): D1 = A @ B1, D2 = A @ B2   (10000x256 @ 256x16, twice)
// Shares A fragments between the two products. 625 M-tiles, 1 N-tile.
// ---------------------------------------------------------------------------
__global__ __launch_bounds__(256) void gemm3_kernel(const float* __restrict__ A,
                                                    const float* __restrict__ B1,
                                                    const float* __restrict__ B2,
                                                    float* __restrict__ D1,
                                                    float* __restrict__ D2) {
    const int lane = threadIdx.x & 31;
    const int wave = threadIdx.x >> 5;
    const int tile = blockIdx.x * 8 + wave;
    if (tile >= 625) return;                 // uniform per-wave exit; EXEC stays full
    const int m0 = tile * 16;

    v8f acc1 = {}, acc2 = {};
#pragma unroll 4
    for (int k0 = 0; k0 < F_HID; k0 += 4) {
        v2f a  = load_a(A, F_HID, m0, k0, lane);
        v2f b1 = load_b(B1, F_OUT, k0, 0, lane);
        v2f b2 = load_b(B2, F_OUT, k0, 0, lane);
        acc1 = wmma4(a, b1, acc1);
        acc2 = wmma4(a, b2, acc2);
    }
    store_c(D1, F_OUT, m0, 0, lane, acc1);
    store_c(D2, F_OUT, m0, 0, lane, acc2);
}

// ---------------------------------------------------------------------------
// BatchNorm statistics: per-column sum and sum-of-squares over the node dim.
// Thread g covers column g%COLS, row slice g/COLS (strided), then atomicAdd.
// ---------------------------------------------------------------------------
template <int COLS>
__global__ void bn_stats_kernel(const float* __restrict__ h,
                                float* __restrict__ sum, float* __restrict__ sumsq) {
    int g = blockIdx.x * blockDim.x + threadIdx.x;
    int col     = g & (COLS - 1);
    int slice   = g / COLS;
    int nslices = (gridDim.x * blockDim.x) / COLS;
    float s = 0.0f, q = 0.0f;
    for (int r = slice; r < N_NODES; r += nslices) {
        float v = h[(size_t)r * COLS + col];
        s += v;
        q += v * v;
    }
    atomicAdd(&sum[col], s);
    atomicAdd(&sumsq[col], q);
}

// BN apply + ReLU (layer 1)
template <int COLS>
__global__ void bn_relu_kernel(float* __restrict__ h,
                               const float* __restrict__ sum, const float* __restrict__ sumsq,
                               const float* __restrict__ gamma, const float* __restrict__ beta,
                               long long total) {
    long long g = (long long)blockIdx.x * blockDim.x + threadIdx.x;
    if (g >= total) return;
    int c = (int)(g & (COLS - 1));
    float mean = sum[c] * (1.0f / N_NODES);
    float var  = sumsq[c] * (1.0f / N_NODES) - mean * mean;
    float v = (h[g] - mean) * rsqrtf(var + BN_EPS) * gamma[c] + beta[c];
    h[g] = fmaxf(v, 0.0f);
}

// agg3[dst] += p[src], whole 16-wide row per thread
__global__ void agg3_scatter_kernel(const float* __restrict__ p,
                                    const int* __restrict__ src,
                                    const int* __restrict__ dst,
                                    float* __restrict__ agg3) {
    int e = blockIdx.x * blockDim.x + threadIdx.x;
    if (e >= N_EDGES) return;
    const float* ps = p + (size_t)src[e] * F_OUT;
    float* ad = agg3 + (size_t)dst[e] * F_OUT;
#pragma unroll
    for (int j = 0; j < F_OUT; j += 4) {
        float4 v = *(const float4*)(ps + j);
        atomicAdd(ad + j + 0, v.x);
        atomicAdd(ad + j + 1, v.y);
        atomicAdd(ad + j + 2, v.z);
        atomicAdd(ad + j + 3, v.w);
    }
}

// t = agg3 * deg_inv + hl   (in place into agg3)
__global__ void combine3_kernel(float* __restrict__ agg3, const float* __restrict__ hl,
                                const float* __restrict__ deg_inv, long long total) {
    long long g = (long long)blockIdx.x * blockDim.x + threadIdx.x;
    if (g < total) agg3[g] = agg3[g] * deg_inv[g / F_OUT] + hl[g];
}

// BN apply + sigmoid / STE (layer 3) -> d_out
__global__ void bn_final_kernel(const float* __restrict__ t,
                                const float* __restrict__ sum, const float* __restrict__ sumsq,
                                const float* __restrict__ gamma, const float* __restrict__ beta,
                                const int* __restrict__ ste,
                                float* __restrict__ out, long long total) {
    long long g = (long long)blockIdx.x * blockDim.x + threadIdx.x;
    if (g >= total) return;
    int c = (int)(g & (F_OUT - 1));
    float mean = sum[c] * (1.0f / N_NODES);
    float var  = sumsq[c] * (1.0f / N_NODES) - mean * mean;
    float v = (t[g] - mean) * rsqrtf(var + BN_EPS) * gamma[c] + beta[c];
    if (*ste)
        out[g] = (v > 0.0f) ? 1.0f : 0.0f;
    else
        out[g] = 1.0f / (1.0f + expf(-v));
}

// ---------------------------------------------------------------------------
// Launcher
// ---------------------------------------------------------------------------
extern "C" void kernel_launch(void* const* d_in, const int* in_sizes, int n_in,
                              void* d_out, int out_size, void* d_ws, size_t ws_size,
                              hipStream_t stream) {
    const float* x        = (const float*)d_in[0];
    const float* W_conv1  = (const float*)d_in[1];
    const float* W_lin1   = (const float*)d_in[2];
    const float* bn1_g    = (const float*)d_in[3];
    const float* bn1_b    = (const float*)d_in[4];
    const float* W_conv3  = (const float*)d_in[5];
    const float* W_lin3   = (const float*)d_in[6];
    const float* bn3_g    = (const float*)d_in[7];
    const float* bn3_b    = (const float*)d_in[8];
    const int*   src      = (const int*)d_in[9];
    const int*   dst      = (const int*)d_in[10];
    const int*   ste      = (const int*)d_in[11];
    float*       out      = (float*)d_out;

    // Workspace layout (floats)
    float* ws = (float*)d_ws;
    const size_t o_deg  = 0;                       // 10000 (padded 10240)
    const size_t o_agg1 = 10240;                   // 2,560,000
    const size_t o_h    = o_agg1 + 2560000;        // 2,560,000
    const size_t o_p    = o_h + 2560000;           //   160,000
    const size_t o_hl   = o_p + 160000;            //   160,000
    const size_t o_agg3 = o_hl + 160000;           //   160,000
    const size_t o_s1   = o_agg3 + 160000;         //       256
    const size_t o_q1   = o_s1 + 256;              //       256
    const size_t o_s3   = o_q1 + 256;              //        16
    const size_t o_q3   = o_s3 + 16;               //        16
    float* deg_inv = ws + o_deg;
    float* agg1    = ws + o_agg1;
    float* h       = ws + o_h;
    float* p       = ws + o_p;
    float* hl      = ws + o_hl;
    float* agg3    = ws + o_agg3;
    float* s1 = ws + o_s1; float* q1 = ws + o_q1;
    float* s3 = ws + o_s3; float* q3 = ws + o_q3;

    // Zero the accumulator regions (graph-capturable async memsets)
    hipMemsetAsync(deg_inv, 0, 10240 * sizeof(float), stream);
    hipMemsetAsync(agg1, 0, 2560000 * sizeof(float), stream);
    hipMemsetAsync(agg3, 0, (160000 + 256 + 256 + 16 + 16) * sizeof(float), stream);

    // Degree -> inverse degree
    deg_accum_kernel<<<(N_EDGES + 255) / 256, 256, 0, stream>>>(dst, deg_inv);
    deg_inv_kernel<<<(N_NODES + 255) / 256, 256, 0, stream>>>(deg_inv);

    // Layer 1 aggregation: agg1 = segsum(x[src] -> dst) * deg_inv
    {
        long long total = (long long)N_EDGES * 64;
        agg1_scatter_kernel<<<(int)((total + 255) / 256), 256, 0, stream>>>(x, src, dst, agg1);
        long long tot2 = (long long)N_NODES * F_IN;
        scale_rows_kernel<F_IN><<<(int)((tot2 + 255) / 256), 256, 0, stream>>>(agg1, deg_inv, tot2);
    }

    // Layer 1 GEMM (WMMA f32): h = agg1 @ W_conv1 + x @ W_lin1
    gemm1_kernel<<<(2500 + 7) / 8, 256, 0, stream>>>(agg1, W_conv1, x, W_lin1, h);

    // BN1 + ReLU
    bn_stats_kernel<F_HID><<<128, 256, 0, stream>>>(h, s1, q1);
    {
        long long tot = (long long)N_NODES * F_HID;
        bn_relu_kernel<F_HID><<<(int)((tot + 255) / 256), 256, 0, stream>>>(h, s1, q1, bn1_g, bn1_b, tot);
    }

    // Layer 3 GEMMs (WMMA f32): p = h @ W_conv3, hl = h @ W_lin3
    gemm3_kernel<<<(625 + 7) / 8, 256, 0, stream>>>(h, W_conv3, W_lin3, p, hl);

    // Layer 3 aggregation + combine: agg3 = segsum(p[src]->dst)*deg_inv + hl
    agg3_scatter_kernel<<<(N_EDGES + 255) / 256, 256, 0, stream>>>(p, src, dst, agg3);
    {
        long long tot = (long long)N_NODES * F_OUT;
        combine3_kernel<<<(int)((tot + 255) / 256), 256, 0, stream>>>(agg3, hl, deg_inv, tot);
    }

    // BN3 + sigmoid / STE -> out
    bn_stats_kernel<F_OUT><<<64, 256, 0, stream>>>(agg3, s3, q3);
    {
        long long tot = (long long)N_NODES * F_OUT;
        bn_final_kernel<<<(int)((tot + 255) / 256), 256, 0, stream>>>(agg3, s3, q3, bn3_g, bn3_b, ste, out, tot);
    }
}